// QuantumLayer_61753039781981
// MI455X (gfx1250) — compile-verified
//
#include <hip/hip_runtime.h>

typedef float v2f __attribute__((ext_vector_type(2)));
typedef float v8f __attribute__((ext_vector_type(8)));

struct cpx { float re, im; };

__device__ __forceinline__ cpx cmul(cpx a, cpx b) {
    return cpx{a.re * b.re - a.im * b.im, a.re * b.im + a.im * b.re};
}
__device__ __forceinline__ cpx cmadd(cpx a, cpx b, cpx acc) {
    return cpx{fmaf(a.re, b.re, fmaf(-a.im, b.im, acc.re)),
               fmaf(a.re, b.im, fmaf( a.im, b.re, acc.im))};
}

struct G2 { cpx m00, m01, m10, m11; };

__device__ __forceinline__ G2 make_u3(const float* p) {
    float st, ct, sp, cp, sl, cl;
    __sincosf(0.5f * p[0], &st, &ct);
    __sincosf(p[1], &sp, &cp);
    __sincosf(p[2], &sl, &cl);
    G2 g;
    g.m00 = cpx{ct, 0.0f};
    g.m01 = cpx{-cl * st, -sl * st};
    g.m10 = cpx{cp * st, sp * st};
    g.m11 = cpx{(cp * cl - sp * sl) * ct, (cp * sl + sp * cl) * ct};
    return g;
}

// row-vector v <- v * G where G acts on wire0 (MSB), index = 2*bit0 + bit1
__device__ __forceinline__ void rowmul_w0(cpx v[4], const G2& g) {
    cpx n0 = cmadd(v[2], g.m10, cmul(v[0], g.m00));
    cpx n1 = cmadd(v[3], g.m10, cmul(v[1], g.m00));
    cpx n2 = cmadd(v[2], g.m11, cmul(v[0], g.m01));
    cpx n3 = cmadd(v[3], g.m11, cmul(v[1], g.m01));
    v[0] = n0; v[1] = n1; v[2] = n2; v[3] = n3;
}
// row-vector v <- v * G where G acts on wire1 (LSB)
__device__ __forceinline__ void rowmul_w1(cpx v[4], const G2& g) {
    cpx n0 = cmadd(v[1], g.m10, cmul(v[0], g.m00));
    cpx n1 = cmadd(v[1], g.m11, cmul(v[0], g.m01));
    cpx n2 = cmadd(v[3], g.m10, cmul(v[2], g.m00));
    cpx n3 = cmadd(v[3], g.m11, cmul(v[2], g.m01));
    v[0] = n0; v[1] = n1; v[2] = n2; v[3] = n3;
}
// column c <- G * c, G on wire0
__device__ __forceinline__ void colmul_w0(cpx c[4], const G2& g) {
    cpx n0 = cmadd(g.m01, c[2], cmul(g.m00, c[0]));
    cpx n1 = cmadd(g.m01, c[3], cmul(g.m00, c[1]));
    cpx n2 = cmadd(g.m11, c[2], cmul(g.m10, c[0]));
    cpx n3 = cmadd(g.m11, c[3], cmul(g.m10, c[1]));
    c[0] = n0; c[1] = n1; c[2] = n2; c[3] = n3;
}
// column c <- G * c, G on wire1
__device__ __forceinline__ void colmul_w1(cpx c[4], const G2& g) {
    cpx n0 = cmadd(g.m01, c[1], cmul(g.m00, c[0]));
    cpx n1 = cmadd(g.m11, c[1], cmul(g.m10, c[0]));
    cpx n2 = cmadd(g.m01, c[3], cmul(g.m00, c[2]));
    cpx n3 = cmadd(g.m11, c[3], cmul(g.m10, c[2]));
    c[0] = n0; c[1] = n1; c[2] = n2; c[3] = n3;
}

__device__ __forceinline__ v8f wmma4(v2f a, v2f b, v8f c) {
    // D(16x16) = A(16x4) * B(4x16) + C   on the matrix pipe
    return __builtin_amdgcn_wmma_f32_16x16x4_f32(false, a, false, b, (short)0, c,
                                                 false, false);
}

__global__ __launch_bounds__(256)
void qlayer_kernel(const float* __restrict__ x, const float* __restrict__ qs,
                   float* __restrict__ out, int nb, int hw) {
    const int lane = (int)(threadIdx.x & 31u);
    const int m    = lane & 15;          // element within group / matrix row
    const bool hi  = lane >= 16;

    const int wave   = (int)((blockIdx.x * blockDim.x + threadIdx.x) >> 5);
    const int nwaves = (int)((gridDim.x * blockDim.x) >> 5);
    const int NG     = (nb * hw) >> 4;                 // groups of 16 elements
    const int gpw    = (NG + nwaves - 1) / nwaves;     // contiguous chunk / wave
    const int gbeg   = wave * gpw;
    const int gend   = min(NG, gbeg + gpw);
    const int groups_per_b = hw >> 4;

    v2f Ab[3][2];                 // realified block unitaries 1..3, A-layout
    v2f ib0, ib1;                 // initial state (col0 of block 0), B-layout
    ib0.x = 0.f; ib0.y = 0.f; ib1.x = 0.f; ib1.y = 0.f;
    int bcur = -1;

    for (int g = gbeg; g < gend; ++g) {
        const int b = g / groups_per_b;        // wave-uniform
        if (b != bcur) {
            bcur = b;
            // ---- block 0: forward-apply to |00> -> initial state column ----
            {
                const float* pb = qs + (size_t)(b * 4 + 0) * 12;
                G2 gA = make_u3(pb + 0), gB = make_u3(pb + 3);
                G2 gC = make_u3(pb + 6), gD = make_u3(pb + 9);
                cpx c4[4] = {cpx{1.f, 0.f}, cpx{0.f, 0.f}, cpx{0.f, 0.f}, cpx{0.f, 0.f}};
                colmul_w0(c4, gA); colmul_w1(c4, gB);
                { cpx t = c4[2]; c4[2] = c4[3]; c4[3] = t; }   // CNOT(0)
                colmul_w0(c4, gC); colmul_w1(c4, gD);
                { cpx t = c4[2]; c4[2] = c4[3]; c4[3] = t; }
                // B layout: lanes<16 carry comps {Re0,Re1},{Im0,Im1}; lanes>=16 {Re2,Re3},{Im2,Im3}
                ib0.x = hi ? c4[2].re : c4[0].re;
                ib0.y = hi ? c4[3].re : c4[1].re;
                ib1.x = hi ? c4[2].im : c4[0].im;
                ib1.y = hi ? c4[3].im : c4[1].im;
            }
            // ---- blocks 1..3: this lane's row of the realified 8x8, padded to 16 ----
            const int   r    = m & 3;
            const bool  isIm = (m & 4) != 0;
            const float msk  = (m < 8) ? 1.0f : 0.0f;
#pragma unroll
            for (int blk = 1; blk <= 3; ++blk) {
                const float* pb = qs + (size_t)(b * 4 + blk) * 12;
                G2 gA = make_u3(pb + 0), gB = make_u3(pb + 3);
                G2 gC = make_u3(pb + 6), gD = make_u3(pb + 9);
                cpx v4[4];
#pragma unroll
                for (int t = 0; t < 4; ++t) v4[t] = cpx{(t == r) ? 1.0f : 0.0f, 0.0f};
                // e_r^T * U = multiply by leftmost factors first (reverse order)
                { cpx t = v4[2]; v4[2] = v4[3]; v4[3] = t; }   // CNOT (j=1)
                rowmul_w1(v4, gD); rowmul_w0(v4, gC);
                { cpx t = v4[2]; v4[2] = v4[3]; v4[3] = t; }   // CNOT (j=0)
                rowmul_w1(v4, gB); rowmul_w0(v4, gA);
                // realify: rows 0..3 = (Re, -Im), rows 4..7 = (Im, Re)
                float row8[8];
#pragma unroll
                for (int k = 0; k < 4; ++k) {
                    row8[k]     = (isIm ? v4[k].im :  v4[k].re) * msk;
                    row8[4 + k] = (isIm ? v4[k].re : -v4[k].im) * msk;
                }
                // A layout 16x4: lanes<16 hold K={4c,4c+1}, lanes>=16 K={4c+2,4c+3}
                v2f a0, a1;
                a0.x = hi ? row8[2] : row8[0];  a0.y = hi ? row8[3] : row8[1];
                a1.x = hi ? row8[6] : row8[4];  a1.y = hi ? row8[7] : row8[5];
                Ab[blk - 1][0] = a0;  Ab[blk - 1][1] = a1;
            }
        }

        // ---- per-element part: 16 elements / wave, lanes L and L+16 share element ----
        const int e = g * 16 + m;
        const float x0 = x[(size_t)e * 2 + 0];
        const float x1 = x[(size_t)e * 2 + 1];
        // RZ diag angles: theta_j = (+-x0 +-x1)/2 by bits; identical each spectrum step
        const float ha = 0.5f * (x0 + x1);
        const float hd = 0.5f * (x0 - x1);
        const float t0 = hi ? hd : -ha;    // comp pair 0 angle (j=0 or j=2)
        const float t1 = hi ? ha : -hd;    // comp pair 1 angle (j=1 or j=3)
        float sn0, cs0, sn1, cs1;
        __sincosf(t0, &sn0, &cs0);
        __sincosf(t1, &sn1, &cs1);

        v2f bc0 = ib0, bc1 = ib1;          // state in WMMA B layout
        v8f d;
#pragma unroll
        for (int i = 0; i < 3; ++i) {
            // phase rotation: (Re,Im) pairs are local to the lane in B layout
            float R0 = bc0.x * cs0 - bc1.x * sn0;
            float I0 = bc1.x * cs0 + bc0.x * sn0;
            float R1 = bc0.y * cs1 - bc1.y * sn1;
            float I1 = bc1.y * cs1 + bc0.y * sn1;
            bc0.x = R0; bc0.y = R1; bc1.x = I0; bc1.y = I1;

            // block unitary on matrix pipe: two K=4 WMMAs accumulate the 8x8 matvec
            v8f c = {0.f, 0.f, 0.f, 0.f, 0.f, 0.f, 0.f, 0.f};
            c = wmma4(Ab[i][0], bc0, c);
            c = wmma4(Ab[i][1], bc1, c);
            d = c;

            if (i < 2) {
                // D (lane L<16 holds all 8 comps of element L) -> B layout
                float r2 = __shfl(d[2], m, 32);
                float r3 = __shfl(d[3], m, 32);
                float r6 = __shfl(d[6], m, 32);
                float r7 = __shfl(d[7], m, 32);
                bc0.x = hi ? r2 : d[0];
                bc0.y = hi ? r3 : d[1];
                bc1.x = hi ? r6 : d[4];
                bc1.y = hi ? r7 : d[5];
            }
        }
        // <Z0> = |s0|^2+|s1|^2-|s2|^2-|s3|^2 (lanes<16 own the rows)
        const float ev = (d[0] * d[0] + d[1] * d[1] + d[4] * d[4] + d[5] * d[5])
                       - (d[2] * d[2] + d[3] * d[3] + d[6] * d[6] + d[7] * d[7]);
        if (!hi) out[(size_t)g * 16 + m] = ev;
    }
}

extern "C" void kernel_launch(void* const* d_in, const int* in_sizes, int n_in,
                              void* d_out, int out_size, void* d_ws, size_t ws_size,
                              hipStream_t stream) {
    (void)n_in; (void)out_size; (void)d_ws; (void)ws_size;
    const float* x  = (const float*)d_in[0];
    const float* qs = (const float*)d_in[1];
    float* out = (float*)d_out;
    const int nb = in_sizes[1] / 48;          // 32: SPECTRUM+1=4, ANSATZ=2, wires=2, 3 params
    const int hw = in_sizes[0] / (nb * 2);    // 16384
    qlayer_kernel<<<512, 256, 0, stream>>>(x, qs, out, nb, hw);
}